// SOM_2010044694719
// MI455X (gfx1250) — compile-verified
//
#include <hip/hip_runtime.h>
#include <hip/hip_bf16.h>

// ---------------------------------------------------------------------------
// SOM training step for MI455X (gfx1250, wave32, WMMA).
//
//  x:       [4096, 256]  f32      (d_in[0])
//  weights: [128, 128, 256] f32   (d_in[1])
//  out:     [128, 128, 256] f32   (d_out)
//
// Stages:
//  1) wn2[n]     = ||w_n||^2                                   (VALU)
//  2) Kmat[i][j] = exp(-(i-j)^2 / 18)   (shared Kr==Kc)        (VALU)
//  3) S          = 0                                           (memset kernel)
//  4) bmu search: argmin_n (wn2[n] - 2 x_b . w_n)   via V_WMMA_F32_16X16X4_F32
//                 (x tile held in 128 VGPRs; 1 VMEM load per WMMA)
//  5) scatter:    S[bmu(b)][d] += x[b][d], S[bmu(b)][256] += 1 (f32 HW atomics)
//  6) convC:      T[r0][c][d]  = sum_c0 Kmat[c][c0] S[r0][c0][d]   (WMMA)
//  7) convR:      num[r][c][d] = sum_r0 Kmat[r][r0] T[r0][c][d]    (WMMA)
//  8) update:     out = w + (LR/B)*(num - num[...,256]*w)
// ---------------------------------------------------------------------------

typedef __attribute__((ext_vector_type(2))) float v2f;
typedef __attribute__((ext_vector_type(8))) float v8f;

#define SOM_B      4096
#define SOM_D      256
#define SOM_R      128
#define SOM_C      128
#define SOM_N      (SOM_R * SOM_C)          // 16384 neurons
#define SOM_D2     272                       // 256 data + 1 count + 15 pad (17 tiles of 16)
#define SOM_INV2S2 (1.0f / 18.0f)            // 1/(2*sigma^2), sigma=3
#define SOM_LRB    (0.1f / 4096.0f)          // LR / B

// ---------------------------------------------------------------- wn2 -------
__global__ __launch_bounds__(256) void som_wn2(const float* __restrict__ w,
                                               float* __restrict__ wn2) {
  const int lane = threadIdx.x & 31;
  const int wave = threadIdx.x >> 5;
  const int n = blockIdx.x * 8 + wave;            // one wave per neuron
  const float* p = w + (size_t)n * SOM_D;
  float s = 0.0f;
#pragma unroll
  for (int k = lane; k < SOM_D; k += 32) {
    float v = p[k];
    s += v * v;
  }
#pragma unroll
  for (int off = 16; off > 0; off >>= 1) s += __shfl_xor(s, off, 32);
  if (lane == 0) wn2[n] = s;
}

// ---------------------------------------------------------------- Kmat ------
__global__ __launch_bounds__(128) void som_kmat(float* __restrict__ Km) {
  const int i = blockIdx.x;                       // 128
  const int j = threadIdx.x;                      // 128
  const float d = (float)(i - j);
  Km[i * SOM_C + j] = expf(-d * d * SOM_INV2S2);
}

// ---------------------------------------------------------------- zero ------
__global__ __launch_bounds__(256) void som_zero(float* __restrict__ p, int n) {
  int i = blockIdx.x * blockDim.x + threadIdx.x;
  int stride = gridDim.x * blockDim.x;
  for (; i < n; i += stride) p[i] = 0.0f;
}

// ------------------------------------------------------- bmu partial --------
// grid (32 sample-groups, 8 neuron-chunks); block = 256 (8 waves x 16 samples)
// D[neuron (M) x sample (N)] accumulated with V_WMMA_F32_16X16X4_F32.
// The wave's x-tile (B operand, 16 samples x 256 d) is loaded ONCE into a
// 64-entry v2f register array (128 VGPRs) and reused across all neuron tiles,
// so the hot loop issues exactly one global_load_b64 per WMMA.
__global__ __launch_bounds__(256) void som_bmu_partial(
    const float* __restrict__ x, const float* __restrict__ w,
    const float* __restrict__ wn2, float* __restrict__ pbest,
    int* __restrict__ pidx) {
  const int lane = threadIdx.x & 31;
  const int wave = threadIdx.x >> 5;
  const int m = lane & 15;            // column within 16-wide tile
  const int h = lane >> 4;            // half-wave select (K pairs / row halves)
  const int sb = (blockIdx.x * 8 + wave) * 16;     // 16 samples per wave
  const int nbase = blockIdx.y * 2048;             // 2048 neurons per chunk

  // B operand (x^T tile): lane holds x[sb+m][k + 2h + {0,1}] -> registers.
  const float* pB = x + (size_t)(sb + m) * SOM_D + 2 * h;
  v2f xr[64];
#pragma unroll
  for (int kk = 0; kk < 64; ++kk) xr[kk] = *(const v2f*)(pB + 4 * kk);

  float best = 3.0e38f;
  int bestIdx = 0;

  for (int nt = 0; nt < 128; ++nt) {              // 128 neuron tiles / chunk
    const int nb = nbase + nt * 16;
    // A operand (w tile): lane holds w[nb+m][k + 2h + {0,1}]
    const float* pA = w + (size_t)(nb + m) * SOM_D + 2 * h;
    if (nt + 1 < 128) {
      // pull next weight tile toward L0/WGP$ (global_prefetch_b8)
      __builtin_prefetch(pA + 16 * SOM_D, 0, 1);
    }
    v8f acc = {};
#pragma unroll
    for (int k = 0; k < 64; ++k) {
      v2f a = *(const v2f*)(pA + 4 * k);
      acc = __builtin_amdgcn_wmma_f32_16x16x4_f32(
          false, a, false, xr[k], (short)0, acc, false, false);
    }
    // lane covers neuron rows nb+8h .. nb+8h+7 for sample sb+m
    const float* pw = wn2 + nb + 8 * h;
#pragma unroll
    for (int v = 0; v < 8; ++v) {
      float score = pw[v] - 2.0f * acc[v];        // ||w||^2 - 2 x.w
      int idx = nb + 8 * h + v;
      if (score < best) { best = score; bestIdx = idx; }
    }
  }
  // merge the two half-rows of each sample column (lane l <-> l+16)
  float ob = __shfl_xor(best, 16, 32);
  int oi = __shfl_xor(bestIdx, 16, 32);
  if (ob < best || (ob == best && oi < bestIdx)) { best = ob; bestIdx = oi; }
  if (lane < 16) {
    pbest[(size_t)blockIdx.y * SOM_B + sb + m] = best;
    pidx[(size_t)blockIdx.y * SOM_B + sb + m] = bestIdx;
  }
}

// ------------------------------------------------------- bmu combine --------
__global__ __launch_bounds__(256) void som_bmu_combine(
    const float* __restrict__ pbest, const int* __restrict__ pidx,
    int* __restrict__ bmu) {
  const int b = blockIdx.x * blockDim.x + threadIdx.x;   // 4096
  float best = 3.0e38f;
  int bi = 0;
#pragma unroll
  for (int c = 0; c < 8; ++c) {     // ascending chunks => ascending idx; '<' keeps first
    float v = pbest[(size_t)c * SOM_B + b];
    int i = pidx[(size_t)c * SOM_B + b];
    if (v < best) { best = v; bi = i; }
  }
  bmu[b] = bi;
}

// ---------------------------------------------------------------- scatter ---
__global__ __launch_bounds__(256) void som_scatter(
    const float* __restrict__ x, const int* __restrict__ bmu,
    float* __restrict__ S) {
  const int b = blockIdx.x;           // 4096 samples
  const int d = threadIdx.x;          // 256
  const int n = bmu[b];
  // native non-returning global_atomic_add_f32
  unsafeAtomicAdd(&S[(size_t)n * SOM_D2 + d], x[(size_t)b * SOM_D + d]);
  if (d == 0) unsafeAtomicAdd(&S[(size_t)n * SOM_D2 + SOM_D], 1.0f);
}

// ---------------------------------------------------------------- convC -----
// T[r0][c][d] = sum_c0 Kmat[c][c0] * S[r0][c0][d]
// grid (r0=128, dtile=17); block 256 = 8 waves (c-tiles of 16)
__global__ __launch_bounds__(256) void som_conv_c(
    const float* __restrict__ Km, const float* __restrict__ S,
    float* __restrict__ T) {
  const int lane = threadIdx.x & 31;
  const int wave = threadIdx.x >> 5;
  const int m = lane & 15, h = lane >> 4;
  const int r0 = blockIdx.x;
  const int db = blockIdx.y * 16;
  const int cb = wave * 16;
  const float* pA = Km + (cb + m) * SOM_C + 2 * h;          // Kmat[c][c0]
  const float* pS = S + (size_t)r0 * (SOM_C * SOM_D2) + db + m;
  v8f acc = {};
#pragma unroll 4
  for (int k = 0; k < SOM_C; k += 4) {
    v2f a = *(const v2f*)(pA + k);
    const int c0 = k + 2 * h;
    v2f b;
    b[0] = pS[(size_t)c0 * SOM_D2];
    b[1] = pS[(size_t)(c0 + 1) * SOM_D2];
    acc = __builtin_amdgcn_wmma_f32_16x16x4_f32(
        false, a, false, b, (short)0, acc, false, false);
  }
  float* pT = T + (size_t)r0 * (SOM_C * SOM_D2) +
              (size_t)(cb + 8 * h) * SOM_D2 + db + m;
#pragma unroll
  for (int v = 0; v < 8; ++v) pT[(size_t)v * SOM_D2] = acc[v];
}

// ---------------------------------------------------------------- convR -----
// num[r][c][d] = sum_r0 Kmat[r][r0] * T[r0][c][d]
// grid (c=128, dtile=17); block 256 = 8 waves (r-tiles of 16)
__global__ __launch_bounds__(256) void som_conv_r(
    const float* __restrict__ Km, const float* __restrict__ T,
    float* __restrict__ num) {
  const int lane = threadIdx.x & 31;
  const int wave = threadIdx.x >> 5;
  const int m = lane & 15, h = lane >> 4;
  const int c = blockIdx.x;
  const int db = blockIdx.y * 16;
  const int rb = wave * 16;
  const float* pA = Km + (rb + m) * SOM_R + 2 * h;          // Kmat[r][r0]
  const float* pT = T + (size_t)c * SOM_D2 + db + m;
  v8f acc = {};
#pragma unroll 4
  for (int k = 0; k < SOM_R; k += 4) {
    v2f a = *(const v2f*)(pA + k);
    const int r0 = k + 2 * h;
    v2f b;
    b[0] = pT[(size_t)r0 * (SOM_C * SOM_D2)];
    b[1] = pT[(size_t)(r0 + 1) * (SOM_C * SOM_D2)];
    acc = __builtin_amdgcn_wmma_f32_16x16x4_f32(
        false, a, false, b, (short)0, acc, false, false);
  }
#pragma unroll
  for (int v = 0; v < 8; ++v) {
    const int r = rb + 8 * h + v;
    num[((size_t)r * SOM_C + c) * SOM_D2 + db + m] = acc[v];
  }
}

// ---------------------------------------------------------------- update ----
__global__ __launch_bounds__(256) void som_update(
    const float* __restrict__ w, const float* __restrict__ num,
    float* __restrict__ out) {
  const int rc = blockIdx.x;          // 16384 neurons
  const int d = threadIdx.x;          // 256
  const float ns = num[(size_t)rc * SOM_D2 + SOM_D];     // neigh_sum[r][c]
  const float wi = w[(size_t)rc * SOM_D + d];
  const float nu = num[(size_t)rc * SOM_D2 + d];
  out[(size_t)rc * SOM_D + d] = wi + SOM_LRB * (nu - ns * wi);
}

// ---------------------------------------------------------------- launch ----
extern "C" void kernel_launch(void* const* d_in, const int* in_sizes, int n_in,
                              void* d_out, int out_size, void* d_ws, size_t ws_size,
                              hipStream_t stream) {
  const float* x = (const float*)d_in[0];     // [4096, 256]
  const float* w = (const float*)d_in[1];     // [128, 128, 256]
  float* out = (float*)d_out;                 // [128, 128, 256]

  char* ws = (char*)d_ws;
  const size_t SZ_S = (size_t)SOM_N * SOM_D2 * sizeof(float);   // 17,825,792
  float* S     = (float*)(ws);                                  // also 'num'
  float* T     = (float*)(ws + SZ_S);
  float* Km    = (float*)(ws + 2 * SZ_S);
  float* wn2   = (float*)(ws + 2 * SZ_S + 64 * 1024);
  int*   bmu   = (int*)  (ws + 2 * SZ_S + 128 * 1024);
  float* pbest = (float*)(ws + 2 * SZ_S + 128 * 1024 + 16 * 1024);
  int*   pidx  = (int*)  (ws + 2 * SZ_S + 128 * 1024 + 16 * 1024 + 128 * 1024);

  // 1) ||w_n||^2
  som_wn2<<<SOM_N / 8, 256, 0, stream>>>(w, wn2);
  // 2) Gaussian kernel matrix (Kr == Kc)
  som_kmat<<<SOM_R, 128, 0, stream>>>(Km);
  // 3) zero histogram S
  som_zero<<<2048, 256, 0, stream>>>(S, SOM_N * SOM_D2);
  // 4) BMU search (fp32 WMMA distance GEMM)
  som_bmu_partial<<<dim3(SOM_B / 128, 8), 256, 0, stream>>>(x, w, wn2, pbest, pidx);
  som_bmu_combine<<<SOM_B / 256, 256, 0, stream>>>(pbest, pidx, bmu);
  // 5) histogram scatter
  som_scatter<<<SOM_B, 256, 0, stream>>>(x, bmu, S);
  // 6,7) separable Gaussian convolution as two WMMA GEMMs
  som_conv_c<<<dim3(SOM_R, SOM_D2 / 16), 256, 0, stream>>>(Km, S, T);
  som_conv_r<<<dim3(SOM_C, SOM_D2 / 16), 256, 0, stream>>>(Km, T, S);  // S reused as num
  // 8) final elementwise update
  som_update<<<SOM_N, 256, 0, stream>>>(w, S, out);
}